// UpSampleModule_76175539962297
// MI455X (gfx1250) — compile-verified
//
#include <hip/hip_runtime.h>
#include <cstdint>
#include <cstddef>

// Problem constants (match reference)
#define B_      16
#define N_      16384
#define BC_     128
#define TIMES_  4

// Tiling
#define TN      128          // points per workgroup tile
#define ROWLEN  132          // LDS row: [px,py,pz,0, 128 features]
#define NSUB    (TN / 16)    // 16-point WMMA sub-tiles per workgroup tile

typedef __attribute__((ext_vector_type(2))) float v2f;
typedef __attribute__((ext_vector_type(8))) float v8f;

// ---- monotonic float <-> uint mapping for atomic max -----------------------
__device__ __forceinline__ unsigned f2key(float f) {
  unsigned u = __float_as_uint(f);
  return (u & 0x80000000u) ? ~u : (u | 0x80000000u);
}
__device__ __forceinline__ float key2f(unsigned k) {
  unsigned u = (k & 0x80000000u) ? (k ^ 0x80000000u) : ~k;
  return __uint_as_float(u);
}

// ---- one 128-out-channel GEMM layer over a TN-point tile -------------------
// D[m, n] = sum_k W[m, k_w] * bufIn[n][k_buf]  + cvec[m]   (optional relu/max)
// WITH_PT: K covers buffer cols 0..4*KSTEPS-1 (col 3 is zero-pad; W cols are
//          0..2 for pt then feature cols shifted by -1).
// else:    K covers buffer cols 4..4+4*KSTEPS-1, W col = buffer col - 4.
template <int KSTEPS, bool WITH_PT, bool RELU, bool DO_STORE, bool DO_MAX>
__device__ __forceinline__ void layer(const float* __restrict__ W, int wstride,
                                      const float* __restrict__ cvec,
                                      const float* bufIn, float* bufOut,
                                      unsigned* __restrict__ maxOut,
                                      int mbase, int lane) {
  const int half = lane >> 4;      // K-half (A/B) and M-half (C/D)
  const int lm   = lane & 15;      // M row (A) / N column (B, C/D)
  const int koff = half * 2;
  const int m    = mbase + lm;

  // A fragments: one float2 per 4-wide K step, held for all N sub-tiles.
  v2f a[KSTEPS];
#pragma unroll
  for (int s = 0; s < KSTEPS; ++s) {
    if (WITH_PT && s == 0) {
      if (half == 0) {
        a[s].x = W[(size_t)m * wstride + 0];
        a[s].y = W[(size_t)m * wstride + 1];
      } else {
        a[s].x = W[(size_t)m * wstride + 2];
        a[s].y = 0.0f;                      // zero-padded K=3 column
      }
    } else {
      const int wc = WITH_PT ? (4 * s + koff - 1) : (4 * s + koff);
      a[s].x = W[(size_t)m * wstride + wc];
      a[s].y = W[(size_t)m * wstride + wc + 1];
    }
  }

  // per-output-channel additive constant (bias + folded broadcast terms)
  float creg[8];
#pragma unroll
  for (int j = 0; j < 8; ++j) creg[j] = cvec[mbase + half * 8 + j];

  float rmax[8];
#pragma unroll
  for (int j = 0; j < 8; ++j) rmax[j] = -3.0e38f;

  for (int nt = 0; nt < NSUB; ++nt) {
    v8f acc = {0.f, 0.f, 0.f, 0.f, 0.f, 0.f, 0.f, 0.f};
    const float* bp = bufIn + (nt * 16 + lm) * ROWLEN + (WITH_PT ? 0 : 4) + koff;
#pragma unroll
    for (int s = 0; s < KSTEPS; ++s) {
      v2f bv = *(const v2f*)(bp + 4 * s);   // ds_load_b64, conflict-free
      acc = __builtin_amdgcn_wmma_f32_16x16x4_f32(false, a[s], false, bv,
                                                  (short)0, acc, false, false);
    }
    float vals[8];
#pragma unroll
    for (int j = 0; j < 8; ++j) {
      float v = acc[j] + creg[j];
      if (RELU) v = fmaxf(v, 0.0f);
      vals[j] = v;
      if (DO_MAX) rmax[j] = fmaxf(rmax[j], v);
    }
    if (DO_STORE) {
      float* o = bufOut + (nt * 16 + lm) * ROWLEN + 4 + mbase + half * 8;
      *(float4*)(o)     = float4{vals[0], vals[1], vals[2], vals[3]};
      *(float4*)(o + 4) = float4{vals[4], vals[5], vals[6], vals[7]};
    }
  }

  if (DO_MAX) {
#pragma unroll
    for (int j = 0; j < 8; ++j) {
      float v = rmax[j];
      v = fmaxf(v, __shfl_xor(v, 1, 32));
      v = fmaxf(v, __shfl_xor(v, 2, 32));
      v = fmaxf(v, __shfl_xor(v, 4, 32));
      v = fmaxf(v, __shfl_xor(v, 8, 32));   // max over the 16-lane half (= over n)
      if (lm == j) atomicMax(maxOut + mbase + half * 8 + j, f2key(v));
    }
  }
}

// ---- final 12-channel layer fused with output write ------------------------
__device__ __forceinline__ void final_layer(const float* __restrict__ w31,
                                            const float* __restrict__ b31,
                                            const float* bufIn,  // acts + pt
                                            float* __restrict__ out,
                                            int b, int nbase, int wave, int lane) {
  const int half = lane >> 4;
  const int lm   = lane & 15;
  const int koff = half * 2;
  const int m    = lm;  // W31 row, valid < 12

  v2f a[32];
#pragma unroll
  for (int s = 0; s < 32; ++s) {
    if (m < 12) {
      a[s].x = w31[m * 128 + 4 * s + koff];
      a[s].y = w31[m * 128 + 4 * s + koff + 1];
    } else {
      a[s].x = 0.0f;
      a[s].y = 0.0f;
    }
  }

  const int nloc = wave * 16 + lm;   // each wave owns one 16-point sub-tile
  v8f acc = {0.f, 0.f, 0.f, 0.f, 0.f, 0.f, 0.f, 0.f};
  const float* bp = bufIn + nloc * ROWLEN + 4 + koff;
#pragma unroll
  for (int s = 0; s < 32; ++s) {
    v2f bv = *(const v2f*)(bp + 4 * s);
    acc = __builtin_amdgcn_wmma_f32_16x16x4_f32(false, a[s], false, bv,
                                                (short)0, acc, false, false);
  }

  const float px = bufIn[nloc * ROWLEN + 0];
  const float py = bufIn[nloc * ROWLEN + 1];
  const float pz = bufIn[nloc * ROWLEN + 2];
  const size_t nglob = (size_t)nbase + nloc;
  float* obase = out + ((size_t)b * N_ + nglob) * (TIMES_ * 3);

#pragma unroll
  for (int j = 0; j < 8; ++j) {
    const int mo = half * 8 + j;     // output channel 0..15, valid < 12
    if (mo < 12) {
      const int t = mo / 3;
      const int k = mo - 3 * t;
      const float pv = (k == 0) ? px : ((k == 1) ? py : pz);
      obase[t * 3 + k] = acc[j] + b31[mo] + pv;  // out[b, n*TIMES+t, k]
    }
  }
}

// ---- fused pass kernel (STAGE = how deep we go this pass) ------------------
template <int STAGE>
__global__ __launch_bounds__(256) void pass_kernel(
    const float* __restrict__ p,
    const float* __restrict__ w10, const float* __restrict__ w11,
    const float* __restrict__ b11, const float* __restrict__ w20,
    const float* __restrict__ w21, const float* __restrict__ b21,
    const float* __restrict__ w30, const float* __restrict__ w31,
    const float* __restrict__ b31, const float* __restrict__ c1,
    const float* __restrict__ c2, const float* __restrict__ c3,
    unsigned* __restrict__ x1max, unsigned* __restrict__ x2max,
    float* __restrict__ out) {
  __shared__ __align__(16) float bufA[TN * ROWLEN];
  __shared__ __align__(16) float bufB[TN * ROWLEN];

  const int b     = blockIdx.y;
  const int nbase = blockIdx.x * TN;
  const int tid   = threadIdx.x;

  // stage point coords into cols 0..3 of both ping-pong buffers
  if (tid < TN) {
    const float* pp = p + ((size_t)b * N_ + nbase + tid) * 3;
    const float px = pp[0], py = pp[1], pz = pp[2];
    float* ra = bufA + tid * ROWLEN;
    float* rb = bufB + tid * ROWLEN;
    ra[0] = px; ra[1] = py; ra[2] = pz; ra[3] = 0.0f;
    rb[0] = px; rb[1] = py; rb[2] = pz; rb[3] = 0.0f;
  }
  __syncthreads();

  const int lane  = tid & 31;
  const int wave  = tid >> 5;
  const int mbase = wave * 16;

  // mlp1 layer0: relu(W10[:, :3] @ pt + c1)            (K = 4 padded)
  layer<1, true, true, true, false>(w10, 131, c1 + b * BC_, bufA, bufB,
                                    (unsigned*)nullptr, mbase, lane);
  __syncthreads();
  // mlp1 layer1: W11 @ h + b11                          (K = 128)
  layer<32, false, false, true, (STAGE == 1)>(w11, 128, b11, bufB, bufA,
                                              x1max + b * BC_, mbase, lane);
  if (STAGE == 1) return;
  __syncthreads();
  // mlp2 layer0: relu(W20[:, :131-map] @ [pt;x1] + c2)  (K = 132 padded)
  layer<33, true, true, true, false>(w20, 259, c2 + b * BC_, bufA, bufB,
                                     (unsigned*)nullptr, mbase, lane);
  __syncthreads();
  layer<32, false, false, true, (STAGE == 2)>(w21, 128, b21, bufB, bufA,
                                              x2max + b * BC_, mbase, lane);
  if (STAGE == 2) return;
  __syncthreads();
  layer<33, true, true, true, false>(w30, 259, c3 + b * BC_, bufA, bufB,
                                     (unsigned*)nullptr, mbase, lane);
  __syncthreads();
  final_layer(w31, b31, bufB, out, b, nbase, wave, lane);
}

// ---- tiny side kernels -----------------------------------------------------
// c1[b,m] = b10[m] + W10[m, 3:131] . gf[b]; also reset atomic-max buffers.
__global__ void init_kernel(const float* __restrict__ w10,
                            const float* __restrict__ b10,
                            const float* __restrict__ gf,
                            float* __restrict__ c1, unsigned* __restrict__ x1m,
                            unsigned* __restrict__ x2m) {
  const int t = blockIdx.x * blockDim.x + threadIdx.x;
  if (t >= B_ * BC_) return;
  const int b = t >> 7, m = t & 127;
  float acc = b10[m];
  const float* wr = w10 + (size_t)m * 131 + 3;
  const float* g  = gf + b * BC_;
  for (int c = 0; c < BC_; ++c) acc += wr[c] * g[c];
  c1[t]  = acc;
  x1m[t] = 0u;   // identity for monotonic-key max (all real keys are larger)
  x2m[t] = 0u;
}

// cout[b,m] = bias[m] + W[m, colOff : colOff+128] . decode(maxu[b])
__global__ void fold_kernel(const float* __restrict__ w, int wstride, int colOff,
                            const float* __restrict__ bias,
                            const unsigned* __restrict__ maxu,
                            float* __restrict__ cout) {
  const int t = blockIdx.x * blockDim.x + threadIdx.x;
  if (t >= B_ * BC_) return;
  const int b = t >> 7, m = t & 127;
  float acc = bias[m];
  const float* wr = w + (size_t)m * wstride + colOff;
  const unsigned* mu = maxu + b * BC_;
  for (int c = 0; c < BC_; ++c) acc += wr[c] * key2f(mu[c]);
  cout[t] = acc;
}

extern "C" void kernel_launch(void* const* d_in, const int* in_sizes, int n_in,
                              void* d_out, int out_size, void* d_ws,
                              size_t ws_size, hipStream_t stream) {
  (void)in_sizes; (void)n_in; (void)out_size; (void)ws_size;
  const float* p   = (const float*)d_in[0];
  const float* gf  = (const float*)d_in[1];
  const float* w10 = (const float*)d_in[2];
  const float* b10 = (const float*)d_in[3];
  const float* w11 = (const float*)d_in[4];
  const float* b11 = (const float*)d_in[5];
  const float* w20 = (const float*)d_in[6];
  const float* b20 = (const float*)d_in[7];
  const float* w21 = (const float*)d_in[8];
  const float* b21 = (const float*)d_in[9];
  const float* w30 = (const float*)d_in[10];
  const float* b30 = (const float*)d_in[11];
  const float* w31 = (const float*)d_in[12];
  const float* b31 = (const float*)d_in[13];
  float* out = (float*)d_out;

  // workspace layout (all re-written every call => graph-replay safe)
  float*    ws  = (float*)d_ws;
  float*    c1  = ws + 0 * B_ * BC_;
  float*    c2  = ws + 1 * B_ * BC_;
  float*    c3  = ws + 2 * B_ * BC_;
  unsigned* x1m = (unsigned*)(ws + 3 * B_ * BC_);
  unsigned* x2m = (unsigned*)(ws + 4 * B_ * BC_);

  const dim3 grid(N_ / TN, B_);
  const dim3 block(256);
  const dim3 sgrid((B_ * BC_ + 127) / 128);
  const dim3 sblock(128);

  init_kernel<<<sgrid, sblock, 0, stream>>>(w10, b10, gf, c1, x1m, x2m);
  pass_kernel<1><<<grid, block, 0, stream>>>(p, w10, w11, b11, w20, w21, b21,
                                             w30, w31, b31, c1, c2, c3, x1m,
                                             x2m, out);
  fold_kernel<<<sgrid, sblock, 0, stream>>>(w20, 259, 131, b20, x1m, c2);
  pass_kernel<2><<<grid, block, 0, stream>>>(p, w10, w11, b11, w20, w21, b21,
                                             w30, w31, b31, c1, c2, c3, x1m,
                                             x2m, out);
  fold_kernel<<<sgrid, sblock, 0, stream>>>(w30, 259, 131, b30, x2m, c3);
  pass_kernel<3><<<grid, block, 0, stream>>>(p, w10, w11, b11, w20, w21, b21,
                                             w30, w31, b31, c1, c2, c3, x1m,
                                             x2m, out);
}